// RPN1D_81535659147632
// MI455X (gfx1250) — compile-verified
//
#include <hip/hip_runtime.h>

typedef float v2f __attribute__((ext_vector_type(2)));
typedef float v4f __attribute__((ext_vector_type(4)));
typedef float v8f __attribute__((ext_vector_type(8)));

#define BATCH 4
#define CCH   512
#define LF    8192
#define ANUM  6
#define NANCH (LF * ANUM)      // 49152
#define KTOT  (CCH * 3)        // 1536 (GEMM K dim: ci*3 + tap, weight memory order)
#define LTILE 64
#define FLDW  72               // feat LDS row stride (floats): cols [l0-4, l0+68)
#define HLDW  66               // H LDS row stride (floats), avoids bank conflicts

__constant__ float c_alen[ANUM] = {2.0f, 4.0f, 6.0f, 9.0f, 13.0f, 18.0f};

__global__ void __launch_bounds__(256)
rpn_fused(const float* __restrict__ feat,
          const float* __restrict__ convw,
          const float* __restrict__ convb,
          const float* __restrict__ objw,
          const float* __restrict__ objb,
          const float* __restrict__ regw,
          const float* __restrict__ regb,
          float* __restrict__ out)
{
    extern __shared__ float smem[];   // max(512*72, 512*66) floats = 147456 B

    const int b    = blockIdx.y;
    const int l0   = blockIdx.x * LTILE;
    const int tid  = threadIdx.x;
    const int wave = tid >> 5;
    const int lane = tid & 31;
    const int half = lane >> 4;   // K half for A/B fragments
    const int mn   = lane & 15;   // M row (A) / N col (B)

    // ---------- Stage feat tile: 512 channels x cols [l0-4, l0+68) ----------
    {
        const float* fb = feat + (size_t)b * CCH * LF;
        v4f* smem4 = (v4f*)smem;
        for (int slot = tid; slot < CCH * (FLDW / 4); slot += 256) {
            const int ci  = slot / (FLDW / 4);
            const int cq  = slot % (FLDW / 4);
            const int col = l0 - 4 + cq * 4;
            v4f v;
            if (col >= 0 && col + 3 < LF) {
                v = *(const v4f*)(fb + (size_t)ci * LF + col);   // aligned b128
            } else {
                #pragma unroll
                for (int j = 0; j < 4; ++j) {
                    const int g = col + j;
                    v[j] = (g >= 0 && g < LF) ? fb[(size_t)ci * LF + g] : 0.0f;
                }
            }
            smem4[slot] = v;
        }
    }
    __syncthreads();

    // ---------- Conv as implicit GEMM via V_WMMA_F32_16X16X4_F32 ----------
    v8f acc[4][4];
    const v8f vzero = {0.f, 0.f, 0.f, 0.f, 0.f, 0.f, 0.f, 0.f};
    #pragma unroll
    for (int i = 0; i < 4; ++i)
        #pragma unroll
        for (int j = 0; j < 4; ++j)
            acc[i][j] = vzero;

    // Per-lane weight row offsets for the 4 co-subtiles (A: M=mn, K=2*half+{0,1})
    int woff[4];
    #pragma unroll
    for (int i = 0; i < 4; ++i)
        woff[i] = (wave * 64 + i * 16 + mn) * KTOT + 2 * half;

    for (int kk = 0; kk < KTOT; kk += 4) {
        const int kd0 = kk + 2 * half;
        const int ci0 = kd0 / 3;
        const int t0  = kd0 - ci0 * 3;
        const int kd1 = kd0 + 1;
        const int ci1 = kd1 / 3;
        const int t1  = kd1 - ci1 * 3;

        v2f Af[4];
        #pragma unroll
        for (int i = 0; i < 4; ++i)
            Af[i] = *(const v2f*)(convw + woff[i] + kk);   // contiguous b64

        // B[kd, l] = feat[ci, l0 + nloc + tap - 1] -> LDS col = nloc + tap + 3
        const int base0 = ci0 * FLDW + t0 + 3 + mn;
        const int base1 = ci1 * FLDW + t1 + 3 + mn;
        v2f Bf[4];
        #pragma unroll
        for (int j = 0; j < 4; ++j) {
            Bf[j][0] = smem[base0 + j * 16];
            Bf[j][1] = smem[base1 + j * 16];
        }

        #pragma unroll
        for (int i = 0; i < 4; ++i)
            #pragma unroll
            for (int j = 0; j < 4; ++j)
                acc[i][j] = __builtin_amdgcn_wmma_f32_16x16x4_f32(
                    false, Af[i], false, Bf[j], (short)0, acc[i][j],
                    false, false);
    }

    // ---------- bias + ReLU, write H tile to LDS (overlay feat region) ----------
    __syncthreads();
    #pragma unroll
    for (int i = 0; i < 4; ++i) {
        #pragma unroll
        for (int r = 0; r < 8; ++r) {
            const int co   = wave * 64 + i * 16 + r + 8 * half; // C layout: M = r + 8*half
            const float bi = convb[co];
            #pragma unroll
            for (int j = 0; j < 4; ++j) {
                float h = acc[i][j][r] + bi;
                h = h > 0.0f ? h : 0.0f;
                smem[co * HLDW + j * 16 + mn] = h;
            }
        }
    }
    __syncthreads();

    // ---------- 1x1 heads: 18 rows x 64 l, reduce over 512 channels ----------
    const int l = tid & 63;
    const int g = tid >> 6;                   // 0..3 (wave-uniform)
    float hacc[5] = {0.f, 0.f, 0.f, 0.f, 0.f};
    const float* wp[5];
    #pragma unroll
    for (int s = 0; s < 5; ++s) {
        const int a = g + 4 * s;              // head row 0..17
        if (a < 6)        wp[s] = objw + a * CCH;
        else if (a < 18)  wp[s] = regw + (a - 6) * CCH;
        else              wp[s] = objw;       // dummy, result discarded
    }
    for (int co = 0; co < CCH; ++co) {
        const float hv = smem[co * HLDW + l];
        #pragma unroll
        for (int s = 0; s < 5; ++s)
            hacc[s] += wp[s][co] * hv;
    }

    const int L = l0 + l;
    float* outReg = out + (size_t)BATCH * NANCH;
    #pragma unroll
    for (int s = 0; s < 5; ++s) {
        const int a = g + 4 * s;
        if (a < 6) {
            out[(size_t)b * NANCH + L * ANUM + a] = hacc[s] + objb[a];
        } else if (a < 18) {
            const int c = a - 6;              // reg channel; flat = l*12 + c
            outReg[(size_t)b * (2 * NANCH) + L * (2 * ANUM) + c] = hacc[s] + regb[c];
        }
    }
}

__global__ void __launch_bounds__(256)
anchors_kernel(float* __restrict__ outA)
{
    const int idx = blockIdx.x * 256 + threadIdx.x;
    if (idx >= NANCH) return;
    const int l = idx / ANUM;
    const int a = idx - l * ANUM;
    const float c = (float)l + 0.5f;
    const float h = 0.5f * c_alen[a];
    outA[2 * idx]     = c - h;
    outA[2 * idx + 1] = c + h;
}

extern "C" void kernel_launch(void* const* d_in, const int* in_sizes, int n_in,
                              void* d_out, int out_size, void* d_ws, size_t ws_size,
                              hipStream_t stream)
{
    const float* feat  = (const float*)d_in[0];
    const float* convw = (const float*)d_in[1];
    const float* convb = (const float*)d_in[2];
    const float* objw  = (const float*)d_in[3];
    const float* objb  = (const float*)d_in[4];
    const float* regw  = (const float*)d_in[5];
    const float* regb  = (const float*)d_in[6];
    float* out = (float*)d_out;

    const size_t smemBytes = (size_t)CCH * FLDW * sizeof(float);   // 147456 B
    (void)hipFuncSetAttribute((const void*)rpn_fused,
                              hipFuncAttributeMaxDynamicSharedMemorySize,
                              (int)smemBytes);

    dim3 grid(LF / LTILE, BATCH);
    rpn_fused<<<grid, 256, smemBytes, stream>>>(feat, convw, convb,
                                                objw, objb, regw, regb, out);

    float* outAnchors = out + (size_t)BATCH * NANCH + (size_t)BATCH * NANCH * 2;
    anchors_kernel<<<(NANCH + 255) / 256, 256, 0, stream>>>(outAnchors);
}